// GNNWeightPredictor_32478542692808
// MI455X (gfx1250) — compile-verified
//
#include <hip/hip_runtime.h>
#include <math.h>

// ---------------------------------------------------------------------------
// GNN weight predictor for MI455X (gfx1250).
// Decoder all-pairs matmul (N^2 x [64x32], ~17.2 GFLOP) runs on
// V_WMMA_F32_16X16X4_F32 to keep exact fp32 semantics while using the
// CDNA5 matrix pipe. Everything else (GATv2 layers) is scatter/softmax
// bound and trivially small (L2-resident working set).
// ---------------------------------------------------------------------------

#define NN 2048
#define NE 65536
#define GSTR 68   // LDS row stride for G tile: bank = (4m+k)%64 -> conflict free

typedef float v2f __attribute__((ext_vector_type(2)));
typedef float v8f __attribute__((ext_vector_type(8)));

// ---- ordered-uint encoding for float atomicMax --------------------------
__device__ __forceinline__ unsigned f2ord(float f) {
  unsigned b = __float_as_uint(f);
  return (b & 0x80000000u) ? ~b : (b | 0x80000000u);
}
__device__ __forceinline__ float ord2f(unsigned u) {
  unsigned b = (u & 0x80000000u) ? (u & 0x7FFFFFFFu) : ~u;
  return __uint_as_float(b);
}

// ---- small utility kernels ----------------------------------------------
__global__ void k_fill_f32(float* p, float v, int n) {
  int i = blockIdx.x * blockDim.x + threadIdx.x;
  if (i < n) p[i] = v;
}
__global__ void k_fill_u32(unsigned* p, unsigned v, int n) {
  int i = blockIdx.x * blockDim.x + threadIdx.x;
  if (i < n) p[i] = v;
}

// out[n, dout] = in[n, din] @ W[din, dout]   (din <= 64, block = dout threads)
__global__ void k_matvec(const float* __restrict__ in, const float* __restrict__ W,
                         float* __restrict__ out, int din, int dout) {
  __shared__ float row[64];
  int n = blockIdx.x, t = threadIdx.x;
  for (int k = t; k < din; k += blockDim.x) row[k] = in[n * din + k];
  __syncthreads();
  float acc = 0.f;
  for (int k = 0; k < din; ++k) acc += row[k] * W[k * dout + t];
  out[n * dout + t] = acc;
}

// ---- GATv2 edge passes ---------------------------------------------------
__global__ void k_edge_score(const int* __restrict__ ei, const float* __restrict__ ea,
                             const float* __restrict__ xl, const float* __restrict__ xr,
                             const float* __restrict__ We, const float* __restrict__ att,
                             float* __restrict__ s, unsigned* __restrict__ smax,
                             int H, int C) {
  int e = blockIdx.x * blockDim.x + threadIdx.x;
  if (e >= NE) return;
  int src = ei[e], dst = ei[NE + e];
  float a = ea[e];
  int HC = H * C;
  const float* xls = xl + src * HC;
  const float* xrd = xr + dst * HC;
  for (int h = 0; h < H; ++h) {
    float acc = 0.f;
    for (int c = 0; c < C; ++c) {
      int o = h * C + c;
      float m = xls[o] + xrd[o] + a * We[o];
      m = (m > 0.f) ? m : 0.2f * m;           // leaky_relu 0.2
      acc += m * att[o];
    }
    s[e * H + h] = acc;
    atomicMax(&smax[dst * H + h], f2ord(acc));
  }
}

__global__ void k_edge_exp(const int* __restrict__ ei, float* __restrict__ s,
                           const unsigned* __restrict__ smax, float* __restrict__ denom,
                           int H) {
  int e = blockIdx.x * blockDim.x + threadIdx.x;
  if (e >= NE) return;
  int dst = ei[NE + e];
  for (int h = 0; h < H; ++h) {
    float ex = expf(s[e * H + h] - ord2f(smax[dst * H + h]));
    s[e * H + h] = ex;
    atomicAdd(&denom[dst * H + h], ex);
  }
}

__global__ void k_edge_aggr(const int* __restrict__ ei, const float* __restrict__ s,
                            const float* __restrict__ denom, const float* __restrict__ xl,
                            float* __restrict__ out, int H, int C) {
  int e = blockIdx.x * blockDim.x + threadIdx.x;
  if (e >= NE) return;
  int src = ei[e], dst = ei[NE + e];
  int HC = H * C;
  for (int h = 0; h < H; ++h) {
    float alpha = s[e * H + h] / denom[dst * H + h];
    for (int c = 0; c < C; ++c) {
      int o = h * C + c;
      atomicAdd(&out[dst * HC + o], alpha * xl[src * HC + o]);
    }
  }
}

// acc[n,D] + bias -> LayerNorm(g,b) -> (optional ELU) -> out[n,D]
__global__ void k_post_ln(const float* __restrict__ acc, const float* __restrict__ bias,
                          const float* __restrict__ g, const float* __restrict__ b,
                          float* __restrict__ out, int D, int do_elu) {
  __shared__ float red[64];
  int n = blockIdx.x, t = threadIdx.x;
  float v = acc[n * D + t] + bias[t];
  red[t] = v; __syncthreads();
  for (int o = D >> 1; o > 0; o >>= 1) { if (t < o) red[t] += red[t + o]; __syncthreads(); }
  float mean = red[0] / (float)D; __syncthreads();
  float d = v - mean;
  red[t] = d * d; __syncthreads();
  for (int o = D >> 1; o > 0; o >>= 1) { if (t < o) red[t] += red[t + o]; __syncthreads(); }
  float var = red[0] / (float)D;
  float y = d * (1.0f / sqrtf(var + 1e-5f)) * g[t] + b[t];
  if (do_elu) y = (y > 0.f) ? y : expm1f(y);
  out[n * D + t] = y;
}

// ---- decoder: all-pairs MLP, g@W2 on V_WMMA_F32_16X16X4_F32 --------------
// One wave handles 16 (i,j) pairs. Block = 8 waves = 128 j's for one i.
__global__ __launch_bounds__(256) void k_decode(
    const float* __restrict__ hs, const float* __restrict__ hd,
    const float* __restrict__ b1, const float* __restrict__ g1, const float* __restrict__ t1,
    const float* __restrict__ W2, const float* __restrict__ b2,
    const float* __restrict__ g2, const float* __restrict__ t2,
    const float* __restrict__ W3, const float* __restrict__ b3,
    float* __restrict__ weights) {
  __shared__ float lds[8 * 16 * GSTR];
  int lane = threadIdx.x & 31;
  int wave = threadIdx.x >> 5;
  int i  = blockIdx.y;
  int jb = blockIdx.x * 128 + wave * 16;
  float* G = &lds[wave * 16 * GSTR];

  int m    = lane & 15;            // row (A) / col (B) index
  int koff = (lane >> 4) * 2;      // K sub-offset: lanes 0-15 -> {0,1}, 16-31 -> {2,3}

  // Preload B fragments: W2 is [64,32] row-major; two N-halves.
  v2f Bf0[16], Bf1[16];
  #pragma unroll
  for (int kk = 0; kk < 16; ++kk) {
    int k0 = kk * 4 + koff;
    v2f b0 = { W2[k0 * 32 + m],      W2[(k0 + 1) * 32 + m] };
    v2f b1v = { W2[k0 * 32 + 16 + m], W2[(k0 + 1) * 32 + 16 + m] };
    Bf0[kk] = b0; Bf1[kk] = b1v;
  }

  // Hoisted per-lane row constants (k = lane and k = lane+32).
  float hsa = hs[i * 64 + lane],      hsb = hs[i * 64 + 32 + lane];
  float b1a = b1[lane],               b1b = b1[lane + 32];
  float g1a = g1[lane],               g1b = g1[lane + 32];
  float t1a = t1[lane],               t1b = t1[lane + 32];

  // Phase 1: build G = leaky(LN(hs[i]+hd[j]+b1)) for 16 pairs, into LDS.
  for (int p = 0; p < 16; ++p) {
    int j = jb + p;
    float a = hsa + hd[j * 64 + lane]      + b1a;
    float b = hsb + hd[j * 64 + 32 + lane] + b1b;
    float sum = a + b, sq = a * a + b * b;
    for (int o = 16; o > 0; o >>= 1) {
      sum += __shfl_xor(sum, o, 32);
      sq  += __shfl_xor(sq,  o, 32);
    }
    float mean = sum * (1.f / 64.f);
    float inv  = 1.0f / sqrtf(sq * (1.f / 64.f) - mean * mean + 1e-5f);
    float ga = (a - mean) * inv * g1a + t1a; ga = (ga > 0.f) ? ga : 0.1f * ga;
    float gb = (b - mean) * inv * g1b + t1b; gb = (gb > 0.f) ? gb : 0.1f * gb;
    G[p * GSTR + lane]      = ga;
    G[p * GSTR + 32 + lane] = gb;
  }

  // Phase 2: [16x64] @ [64x32] via 16 chained f32 WMMAs per N-half.
  v8f c0 = {0.f, 0.f, 0.f, 0.f, 0.f, 0.f, 0.f, 0.f};
  v8f c1 = {0.f, 0.f, 0.f, 0.f, 0.f, 0.f, 0.f, 0.f};
  #pragma unroll
  for (int kk = 0; kk < 16; ++kk) {
    int k0 = kk * 4 + koff;
    v2f av = { G[m * GSTR + k0], G[m * GSTR + k0 + 1] };
    c0 = __builtin_amdgcn_wmma_f32_16x16x4_f32(false, av, false, Bf0[kk],
                                               (short)0, c0, false, false);
    c1 = __builtin_amdgcn_wmma_f32_16x16x4_f32(false, av, false, Bf1[kk],
                                               (short)0, c1, false, false);
  }

  // Phase 3: spill C to LDS (reuse G region), row stride 33.
  float* Cb = G;
  int rbase = (lane >> 4) * 8;     // C layout: VGPR r -> row r (+8 for hi lanes)
  #pragma unroll
  for (int r = 0; r < 8; ++r) {
    Cb[(rbase + r) * 33 + m]      = c0[r];
    Cb[(rbase + r) * 33 + 16 + m] = c1[r];
  }
  __syncthreads();

  // Phase 4: per-pair LN(32) + leaky + W3 dot + sigmoid.
  if (lane < 16) {
    int p = lane;
    float v[32];
    float sum = 0.f, sq = 0.f;
    #pragma unroll
    for (int k = 0; k < 32; ++k) {
      float xv = Cb[p * 33 + k] + b2[k];
      v[k] = xv; sum += xv; sq += xv * xv;
    }
    float mean = sum * (1.f / 32.f);
    float inv  = 1.0f / sqrtf(sq * (1.f / 32.f) - mean * mean + 1e-5f);
    float acc = 0.f;
    #pragma unroll
    for (int k = 0; k < 32; ++k) {
      float y = (v[k] - mean) * inv * g2[k] + t2[k];
      y = (y > 0.f) ? y : 0.1f * y;
      acc += y * W3[k];
    }
    acc += b3[0];
    weights[(size_t)i * NN + jb + p] = 1.0f / (1.0f + expf(-acc));
  }
}

// ---------------------------------------------------------------------------
// Host side
// ---------------------------------------------------------------------------
static void run_conv(hipStream_t st, const float* in, int din,
                     const int* ei, const float* ea,
                     const float* Wl, const float* Wr, const float* We,
                     const float* att, const float* bconv,
                     const float* lng, const float* lnb, int H, int C,
                     float* xl, float* xr, float* sbuf, unsigned* smax,
                     float* denom, float* out, int elu) {
  int HC = H * C;
  k_matvec<<<NN, HC, 0, st>>>(in, Wl, xl, din, HC);
  k_matvec<<<NN, HC, 0, st>>>(in, Wr, xr, din, HC);
  k_fill_u32<<<(NN * H + 255) / 256, 256, 0, st>>>(smax, 0u, NN * H);
  k_fill_f32<<<(NN * H + 255) / 256, 256, 0, st>>>(denom, 0.f, NN * H);
  k_fill_f32<<<(NN * HC + 255) / 256, 256, 0, st>>>(out, 0.f, NN * HC);
  k_edge_score<<<NE / 256, 256, 0, st>>>(ei, ea, xl, xr, We, att, sbuf, smax, H, C);
  k_edge_exp<<<NE / 256, 256, 0, st>>>(ei, sbuf, smax, denom, H);
  k_edge_aggr<<<NE / 256, 256, 0, st>>>(ei, sbuf, denom, xl, out, H, C);
  k_post_ln<<<NN, HC, 0, st>>>(out, bconv, lng, lnb, out, HC, elu);
}

extern "C" void kernel_launch(void* const* d_in, const int* in_sizes, int n_in,
                              void* d_out, int out_size, void* d_ws, size_t ws_size,
                              hipStream_t stream) {
  (void)in_sizes; (void)n_in; (void)out_size; (void)ws_size;
  // setup_inputs() insertion order, params dict flattened recursively in order.
  const float* x     = (const float*)d_in[0];
  const int*   ei    = (const int*)  d_in[1];
  const float* ea    = (const float*)d_in[2];
  const float* c1Wl  = (const float*)d_in[3];
  const float* c1Wr  = (const float*)d_in[4];
  const float* c1We  = (const float*)d_in[5];
  const float* c1att = (const float*)d_in[6];
  const float* c1b   = (const float*)d_in[7];
  const float* c2Wl  = (const float*)d_in[8];
  const float* c2Wr  = (const float*)d_in[9];
  const float* c2We  = (const float*)d_in[10];
  const float* c2att = (const float*)d_in[11];
  const float* c2b   = (const float*)d_in[12];
  const float* c3Wl  = (const float*)d_in[13];
  const float* c3Wr  = (const float*)d_in[14];
  const float* c3We  = (const float*)d_in[15];
  const float* c3att = (const float*)d_in[16];
  const float* c3b   = (const float*)d_in[17];
  const float* n1g = (const float*)d_in[18], *n1b = (const float*)d_in[19];
  const float* n2g = (const float*)d_in[20], *n2b = (const float*)d_in[21];
  const float* n3g = (const float*)d_in[22], *n3b = (const float*)d_in[23];
  const float* dW1 = (const float*)d_in[24], *db1 = (const float*)d_in[25];
  const float* l1g = (const float*)d_in[26], *l1b = (const float*)d_in[27];
  const float* dW2 = (const float*)d_in[28], *db2 = (const float*)d_in[29];
  const float* l2g = (const float*)d_in[30], *l2b = (const float*)d_in[31];
  const float* dW3 = (const float*)d_in[32], *db3 = (const float*)d_in[33];

  float* ws    = (float*)d_ws;
  float* xl    = ws;                    // [N,64]  (later: hs)
  float* xr    = xl + NN * 64;          // [N,64]  (later: hd)
  float* h1    = xr + NN * 64;          // [N,64]
  float* h2    = h1 + NN * 64;          // [N,64]
  float* sbuf  = h2 + NN * 64;          // [E,4]
  unsigned* smax = (unsigned*)(sbuf + NE * 4);   // [N,4]
  float* denom = (float*)(smax + NN * 4);        // [N,4]

  float* weights = (float*)d_out;                // [N,N]
  float* emb     = weights + (size_t)NN * NN;    // [N,32]

  // conv1 (din=7, H=4, C=16) -> ELU(LN) -> h1
  run_conv(stream, x, 7, ei, ea, c1Wl, c1Wr, c1We, c1att, c1b, n1g, n1b,
           4, 16, xl, xr, sbuf, smax, denom, h1, 1);
  // conv2 (din=64) -> ELU(LN) -> h2
  run_conv(stream, h1, 64, ei, ea, c2Wl, c2Wr, c2We, c2att, c2b, n2g, n2b,
           4, 16, xl, xr, sbuf, smax, denom, h2, 1);
  // conv3 (din=64, H=1, C=32) -> LN -> emb (written straight into d_out tail)
  run_conv(stream, h2, 64, ei, ea, c3Wl, c3Wr, c3We, c3att, c3b, n3g, n3b,
           1, 32, xl, xr, sbuf, smax, denom, emb, 0);

  // decoder: hs = emb @ W1[:32], hd = emb @ W1[32:]
  k_matvec<<<NN, 64, 0, stream>>>(emb, dW1, xl, 32, 64);
  k_matvec<<<NN, 64, 0, stream>>>(emb, dW1 + 32 * 64, xr, 32, 64);

  dim3 grid(NN / 128, NN);
  k_decode<<<grid, 256, 0, stream>>>(xl, xr, db1, l1g, l1b, dW2, db2,
                                     l2g, l2b, dW3, db3, weights);
}